// SimpleX_85426899517964
// MI455X (gfx1250) — compile-verified
//
#include <hip/hip_runtime.h>
#include <hip/hip_bf16.h>

#define AS1 __attribute__((address_space(1)))
#define AS3 __attribute__((address_space(3)))

#ifndef __has_builtin
#define __has_builtin(x) 0
#endif

#if __has_builtin(__builtin_amdgcn_global_load_async_to_lds_b32)
#define HAVE_ASYNC_LDS 1
#else
#define HAVE_ASYNC_LDS 0
#endif
#if __has_builtin(__builtin_amdgcn_s_wait_asynccnt)
#define HAVE_WAIT_ASYNC 1
#else
#define HAVE_WAIT_ASYNC 0
#endif

typedef __attribute__((ext_vector_type(2))) float v2f;
typedef __attribute__((ext_vector_type(8))) float v8f;

static constexpr int H = 64;   // hidden size (reference)

// ---------------------------------------------------------------------------
// helpers
// ---------------------------------------------------------------------------
template <int N>
__device__ inline void wait_async() {
#if HAVE_WAIT_ASYNC
    __builtin_amdgcn_s_wait_asynccnt(N);
#elif HAVE_ASYNC_LDS
    asm volatile("s_wait_asynccnt %0" ::"i"(N) : "memory");
#endif
}

__device__ inline void stage_row_async(const float* g, float* lds_slot) {
#if HAVE_ASYNC_LDS
    // per-lane gather: each lane copies one float from global g into LDS at lds_slot (tracked with ASYNCcnt)
    __builtin_amdgcn_global_load_async_to_lds_b32(
        (AS1 int*)(unsigned long long)g,
        (AS3 int*)(unsigned int)(unsigned long long)lds_slot,
        0, 0);
#else
    *lds_slot = *g;   // synchronous fallback
#endif
}

__device__ inline float wave_allreduce_sum(float x) {
#pragma unroll
    for (int m = 16; m >= 1; m >>= 1) x += __shfl_xor(x, m, 32);
    return x;
}

// ---------------------------------------------------------------------------
// Kernel 1: exclusive prefix sums of size[] and target_size[] (single block)
// ---------------------------------------------------------------------------
__global__ void scan_kernel(const int* __restrict__ size,
                            const int* __restrict__ tsize, int nb,
                            int* __restrict__ seg_off, int* __restrict__ tgt_off) {
    __shared__ int part[256];
    const int tid = threadIdx.x, nth = blockDim.x;
    const int chunk = (nb + nth - 1) / nth;
    for (int pass = 0; pass < 2; ++pass) {
        const int* src = pass ? tsize : size;
        int* dst       = pass ? tgt_off : seg_off;
        const int lo = tid * chunk;
        int s = 0;
        for (int j = 0; j < chunk; ++j) { int i = lo + j; if (i < nb) s += src[i]; }
        part[tid] = s;
        __syncthreads();
        for (int o = 1; o < nth; o <<= 1) {
            int v = (tid >= o) ? part[tid - o] : 0;
            __syncthreads();
            part[tid] += v;
            __syncthreads();
        }
        int run = (tid == 0) ? 0 : part[tid - 1];
        for (int j = 0; j < chunk; ++j) {
            int i = lo + j;
            if (i < nb) { dst[i] = run; run += src[i]; }
        }
        if (tid == nth - 1) dst[nb] = part[nth - 1];
        __syncthreads();
    }
}

// ---------------------------------------------------------------------------
// Kernel 2: target index -> group id map
// ---------------------------------------------------------------------------
__global__ void t2g_kernel(const int* __restrict__ tgt_off, int* __restrict__ t2g,
                           int nb, int T) {
    int b = blockIdx.x * blockDim.x + threadIdx.x;
    if (b < nb) {
        int s = tgt_off[b], e = tgt_off[b + 1];
        for (int j = s; j < e && j < T; ++j) t2g[j] = b;
    }
}

// ---------------------------------------------------------------------------
// Kernel 3: per-group embedding (64 threads = one column each).
// Item rows are prefetched into an LDS double buffer with async global->LDS
// loads (ASYNCcnt pipelining), then blended with the user row, centered and
// L2-normalized. Result: embn[b][64].
// ---------------------------------------------------------------------------
__global__ void group_emb_kernel(const int* __restrict__ user,
                                 const int* __restrict__ item,
                                 const int* __restrict__ size,
                                 const float* __restrict__ uw,
                                 const float* __restrict__ iw,
                                 const int* __restrict__ seg_off,
                                 float* __restrict__ embn, int nb) {
    const int b = blockIdx.x;
    if (b >= nb) return;
    const int tid = threadIdx.x;            // 0..63
    const int lane = tid & 31, wv = tid >> 5;
    const int off = seg_off[b];
    const int n = size[b];

    __shared__ float buf[2][H];
    __shared__ float red[2];

    float acc = 0.0f;
    if (n > 0) {
        // stage row 0
        stage_row_async(iw + (long long)item[off] * H + tid, &buf[0][tid]);
        for (int s = 0; s < n; ++s) {
            if (s + 1 < n) {
                stage_row_async(iw + (long long)item[off + s + 1] * H + tid,
                                &buf[(s + 1) & 1][tid]);
                wait_async<1>();   // row s landed; row s+1 still in flight
            } else {
                wait_async<0>();
            }
            acc += buf[s & 1][tid];
        }
    }

    const float inv_n = 1.0f / ((float)n + 1e-6f);
    const long long li = off + (n > 0 ? n : 1) - 1;
    const float uv = uw[(long long)user[li] * H + tid];
    const float e = 0.5f * uv + 0.5f * (acc * inv_n);

    // mean over the 64 columns (2 waves)
    float s = wave_allreduce_sum(e);
    if (lane == 0) red[wv] = s;
    __syncthreads();
    const float mean = (red[0] + red[1]) * (1.0f / (float)H);
    const float c = e - mean;
    float q = wave_allreduce_sum(c * c);
    __syncthreads();
    if (lane == 0) red[wv] = q;
    __syncthreads();
    const float inv = 1.0f / fmaxf(sqrtf(red[0] + red[1]), 1e-12f);
    embn[(size_t)b * H + tid] = c * inv;
}

// ---------------------------------------------------------------------------
// Kernel 4: normalize target rows into LDS, then per-wave WMMA f32 16x16x4
// tiles: A = 16 group-embedding rows (16x64), B = 16 target rows as columns
// (64x16); simplex[t] = diag(A x B).
// 256 threads = 8 waves, 128 targets per block.
// ---------------------------------------------------------------------------
__global__ void dot_kernel(const int* __restrict__ target_item,
                           const float* __restrict__ iw,
                           const float* __restrict__ embn,
                           const int* __restrict__ t2g,
                           float* __restrict__ out, int T) {
    constexpr int TPB = 128;           // targets per block
    constexpr int ROWP = H + 1;        // LDS row pitch (pad: no bank conflicts)
    __shared__ float smem[TPB * ROWP];

    const int lane = threadIdx.x & 31;
    const int w = threadIdx.x >> 5;    // wave id 0..7
    const int tb = blockIdx.x * TPB;

    // ---- phase 1: center + L2-normalize 16 target rows per wave into LDS ----
#pragma unroll 1
    for (int i = 0; i < 16; ++i) {
        const int r = w * 16 + i;
        int tt = tb + r;
        if (tt >= T) tt = T - 1;                     // clamp (tail safety)
        const float* row = iw + (size_t)target_item[tt] * H;
        const float v0 = row[2 * lane];
        const float v1 = row[2 * lane + 1];
        const float mean = wave_allreduce_sum(v0 + v1) * (1.0f / (float)H);
        const float c0 = v0 - mean, c1 = v1 - mean;
        const float q = wave_allreduce_sum(c0 * c0 + c1 * c1);
        const float inv = 1.0f / fmaxf(sqrtf(q), 1e-12f);
        smem[r * ROWP + 2 * lane]     = c0 * inv;
        smem[r * ROWP + 2 * lane + 1] = c1 * inv;
    }
    __syncthreads();

    // ---- phase 2: WMMA tile of 16 targets per wave ----
    const int t0 = tb + w * 16;
    const int m = lane & 15;           // row (A) / col (B) index in tile
    const int kh = lane >> 4;          // K half: lanes 0-15 -> K{0,1}, 16-31 -> K{2,3}
    int tm = t0 + m;
    if (tm >= T) tm = T - 1;
    const float* arow = embn + (size_t)t2g[tm] * H;          // group embedding row
    const float* brow = &smem[(w * 16 + m) * ROWP];          // normalized target row

    v8f c = {};
#pragma unroll
    for (int kk = 0; kk < 16; ++kk) {
        const int kb = 4 * kk + 2 * kh;
        v2f a; a.x = arow[kb]; a.y = arow[kb + 1];
        v2f bb; bb.x = brow[kb]; bb.y = brow[kb + 1];
        // D = A(16x4,f32) x B(4x16,f32) + C  -- full-rate f32 matrix pipe
        c = __builtin_amdgcn_wmma_f32_16x16x4_f32(false, a, false, bb,
                                                  (short)0, c, false, false);
    }

    // diagonal extraction: D[mm][mm] lives at VGPR mm&7, lane mm (mm<8) / mm+16
    float r = 0.0f;
#pragma unroll
    for (int mm = 0; mm < 16; ++mm) {
        const int srcLane = (mm < 8) ? mm : (mm + 16);
        const float v = __shfl(c[mm & 7], srcLane, 32);
        if (lane == mm) r = v;
    }
    if (lane < 16 && (t0 + lane) < T) out[t0 + lane] = r;
}

// ---------------------------------------------------------------------------
// host launcher
// ---------------------------------------------------------------------------
extern "C" void kernel_launch(void* const* d_in, const int* in_sizes, int n_in,
                              void* d_out, int out_size, void* d_ws, size_t ws_size,
                              hipStream_t stream) {
    const int* user        = (const int*)d_in[0];
    const int* item        = (const int*)d_in[1];
    const int* target_item = (const int*)d_in[2];
    const int* size        = (const int*)d_in[3];
    const int* tsize       = (const int*)d_in[4];
    const float* uw        = (const float*)d_in[5];
    const float* iw        = (const float*)d_in[6];
    float* out             = (float*)d_out;

    const int B = in_sizes[3];
    const int T = in_sizes[2];

    // workspace carve-out (256B aligned): seg_off, tgt_off, t2g, embn
    char* w = (char*)d_ws;
    auto carve = [&](size_t bytes) {
        void* p = (void*)w;
        w += (bytes + 255) & ~(size_t)255;
        return p;
    };
    int*   seg_off = (int*)carve((size_t)(B + 1) * sizeof(int));
    int*   tgt_off = (int*)carve((size_t)(B + 1) * sizeof(int));
    int*   t2g     = (int*)carve((size_t)T * sizeof(int));
    float* embn    = (float*)carve((size_t)B * H * sizeof(float));

    scan_kernel<<<1, 256, 0, stream>>>(size, tsize, B, seg_off, tgt_off);
    t2g_kernel<<<(B + 255) / 256, 256, 0, stream>>>(tgt_off, t2g, B, T);
    group_emb_kernel<<<B, H, 0, stream>>>(user, item, size, uw, iw, seg_off, embn, B);
    dot_kernel<<<(T + 127) / 128, 256, 0, stream>>>(target_item, iw, embn, t2g, out, T);
}